// BaseBidirectionalAttention_79173427134888
// MI455X (gfx1250) — compile-verified
//
#include <hip/hip_runtime.h>
#include <hip/hip_bf16.h>

typedef __attribute__((ext_vector_type(2))) float v2f;
typedef __attribute__((ext_vector_type(8))) float v8f;

constexpr int Bb = 16, C = 2048, Q = 128, E = 256;
constexpr int QPAD = 260;   // question LDS row stride (floats): 4-bank skew, 16B-aligned rows
constexpr int SPAD = 132;   // sim LDS row stride (floats): 4-bank skew, 8B-aligned rows
constexpr int WAVES = 4;    // waves per workgroup (wave32)
constexpr int ROWS_PER_WG = WAVES * 16;   // 64 context rows per workgroup

__device__ __forceinline__ v8f wmma_f32_4(v2f a, v2f b, v8f c) {
    // D = A(16x4,f32) * B(4x16,f32) + C(16x16,f32), full f32 precision
    return __builtin_amdgcn_wmma_f32_16x16x4_f32(false, a, false, b, (short)0, c, false, false);
}

// ---------------------------------------------------------------------------
// Kernel A: qw[b,q] = question[b,q,:] . w_question
// ---------------------------------------------------------------------------
__global__ void qw_kernel(const float* __restrict__ question,
                          const float* __restrict__ w_question,
                          float* __restrict__ qw) {
    const int b = blockIdx.x;
    const int q = threadIdx.x;
    const float* r = question + ((size_t)b * Q + q) * E;
    float s = 0.f;
    for (int e = 0; e < E; e += 4) {
        float4 v = *(const float4*)(r + e);
        s += v.x * w_question[e]     + v.y * w_question[e + 1]
           + v.z * w_question[e + 2] + v.w * w_question[e + 3];
    }
    qw[b * Q + q] = s;
}

// ---------------------------------------------------------------------------
// Kernel B: per (b, 64-row context tile):
//   sim = ctx @ (wm*ques)^T + cw + qw   (WMMA f32 16x16x4)
//   P   = softmax_q(sim); mrow = rowmax(sim)
//   c2q = P @ ques                      (WMMA f32 16x16x4)
//   out[:, E:2E]  = c2q ;  out[:, 2E:3E] = ctx * c2q
// ---------------------------------------------------------------------------
__global__ void __launch_bounds__(WAVES * 32) bidaf_main(
    const float* __restrict__ context,
    const float* __restrict__ question,
    const float* __restrict__ w_context,
    const float* __restrict__ w_multiple,
    const float* __restrict__ qw,
    float* __restrict__ mrow,
    float* __restrict__ out)
{
    extern __shared__ float smem[];
    float* qlds   = smem;                        // Q * QPAD
    float* simlds = qlds + Q * QPAD;             // WAVES * 16 * SPAD
    float* wmlds  = simlds + WAVES * 16 * SPAD;  // E
    float* wclds  = wmlds + E;                   // E
    float* qwlds  = wclds + E;                   // Q
    float* cwlds  = qwlds + Q;                   // WAVES * 16

    const int b    = blockIdx.y;
    const int c0b  = blockIdx.x * ROWS_PER_WG;
    const int tid  = threadIdx.x;
    const int wave = tid >> 5;      // wave32
    const int lane = tid & 31;
    const int half = lane >> 4;     // 0: lanes 0-15, 1: lanes 16-31
    const int n    = lane & 15;

    // ---- stage question[b] row-major + small vectors into LDS ----
    const float* qb = question + (size_t)b * Q * E;
    for (int f = tid; f < (Q * E) / 4; f += WAVES * 32) {
        const int q = (f * 4) / E, e = (f * 4) % E;
        *(float4*)(qlds + q * QPAD + e) = *(const float4*)(qb + q * E + e);
    }
    for (int e = tid; e < E; e += WAVES * 32) { wmlds[e] = w_multiple[e]; wclds[e] = w_context[e]; }
    for (int q = tid; q < Q; q += WAVES * 32) qwlds[q] = qw[b * Q + q];
    __syncthreads();

    const int c0 = c0b + wave * 16;
    const float* ctile = context + ((size_t)b * C + c0) * E;

    // ---- cw[m] = ctx row . w_context (lane pair split over E, combine via xor-16) ----
    {
        const float* r = ctile + n * E + half * (E / 2);
        const float* w = wclds + half * (E / 2);
        float s = 0.f;
        for (int e = 0; e < E / 2; e += 4) {
            float4 v = *(const float4*)(r + e);
            s += v.x * w[e] + v.y * w[e + 1] + v.z * w[e + 2] + v.w * w[e + 3];
        }
        s += __shfl_xor(s, 16, 32);
        if (half == 0) cwlds[wave * 16 + n] = s;
    }
    __syncthreads();

    // ---- similarity GEMM: 8 q-tiles of 16, K=E=256 in steps of 4 ----
    v8f acc[8] = {};
    for (int t = 0; t < 64; ++t) {
        const int e = 4 * t + 2 * half;                 // this lane's K pair
        const float2 av = *(const float2*)(ctile + n * E + e);   // A[m=n][e..e+1]
        v2f a; a.x = av.x; a.y = av.y;
        const float wm0 = wmlds[e], wm1 = wmlds[e + 1];
#pragma unroll
        for (int j = 0; j < 8; ++j) {
            const float2 qv = *(const float2*)(qlds + (j * 16 + n) * QPAD + e);
            v2f bf; bf.x = qv.x * wm0; bf.y = qv.y * wm1;  // B[e][q] = wm[e]*ques[q][e]
            acc[j] = wmma_f32_4(a, bf, acc[j]);
        }
    }

    // ---- D-layout -> LDS with row/col biases added ----
    float* mysim = simlds + wave * 16 * SPAD;
#pragma unroll
    for (int j = 0; j < 8; ++j)
#pragma unroll
        for (int v = 0; v < 8; ++v) {
            const int m = v + 8 * half;      // D: VGPR v, lane half -> row
            const int q = j * 16 + n;        // D: N = lane&15
            mysim[m * SPAD + q] = acc[j][v] + cwlds[wave * 16 + m] + qwlds[q];
        }
    __syncthreads();

    // ---- row softmax over q (lane pair owns half a row), save rowmax ----
    {
        float* rowp = mysim + n * SPAD + half * (Q / 2);
        float mx = -3.4e38f;
        for (int i = 0; i < Q / 2; ++i) mx = fmaxf(mx, rowp[i]);
        mx = fmaxf(mx, __shfl_xor(mx, 16, 32));
        float s = 0.f;
        for (int i = 0; i < Q / 2; ++i) { const float ev = __expf(rowp[i] - mx); rowp[i] = ev; s += ev; }
        s += __shfl_xor(s, 16, 32);
        const float inv = 1.f / s;
        for (int i = 0; i < Q / 2; ++i) rowp[i] *= inv;
        if (half == 0) mrow[(size_t)b * C + c0 + n] = mx;
    }
    __syncthreads();

    // ---- c2q = P @ question, E processed in 4 chunks of 64 to bound VGPRs ----
    for (int ec = 0; ec < 4; ++ec) {
        v8f acc2[4] = {};
        for (int t = 0; t < 32; ++t) {
            const int kp = 4 * t + 2 * half;
            const float2 av = *(const float2*)(mysim + n * SPAD + kp);  // A[m=n][kp..kp+1]
            v2f a; a.x = av.x; a.y = av.y;
#pragma unroll
            for (int j = 0; j < 4; ++j) {
                const int e = ec * 64 + j * 16 + n;
                v2f bf;
                bf.x = qlds[kp * QPAD + e];          // B[q=kp][e]
                bf.y = qlds[(kp + 1) * QPAD + e];
                acc2[j] = wmma_f32_4(a, bf, acc2[j]);
            }
        }
#pragma unroll
        for (int j = 0; j < 4; ++j)
#pragma unroll
            for (int v = 0; v < 8; ++v) {
                const int m = v + 8 * half;
                const int e = ec * 64 + j * 16 + n;
                const size_t row = (size_t)b * C + c0 + m;
                const float c2qv = acc2[j][v];
                const float cv   = context[row * E + e];
                float* orow = out + row * (4 * E);
                orow[E + e]     = c2qv;          // segment 1: c2q
                orow[2 * E + e] = cv * c2qv;     // segment 2: ctx * c2q
            }
    }
}

// ---------------------------------------------------------------------------
// Kernel C: per b — softmax over C of row-max, then q2c[b,e] = sum_c w[c]*ctx
// ---------------------------------------------------------------------------
__global__ void q2c_kernel(const float* __restrict__ context,
                           const float* __restrict__ mrow,
                           float* __restrict__ q2c) {
    __shared__ float w[C];
    __shared__ float red[256];
    const int b = blockIdx.x, tid = threadIdx.x;

    float mx = -3.4e38f;
    for (int c = tid; c < C; c += 256) { const float v = mrow[b * C + c]; w[c] = v; mx = fmaxf(mx, v); }
    red[tid] = mx; __syncthreads();
    for (int s = 128; s > 0; s >>= 1) { if (tid < s) red[tid] = fmaxf(red[tid], red[tid + s]); __syncthreads(); }
    mx = red[0]; __syncthreads();

    float sum = 0.f;
    for (int c = tid; c < C; c += 256) { const float ev = __expf(w[c] - mx); w[c] = ev; sum += ev; }
    red[tid] = sum; __syncthreads();
    for (int s = 128; s > 0; s >>= 1) { if (tid < s) red[tid] += red[tid + s]; __syncthreads(); }
    const float inv = 1.f / red[0];
    __syncthreads();

    // tid == e (E == 256); coalesced context reads, LDS broadcast of w[c]
    float acc = 0.f;
    for (int c = 0; c < C; ++c) acc += w[c] * context[((size_t)b * C + c) * E + tid];
    q2c[b * E + tid] = acc * inv;
}

// ---------------------------------------------------------------------------
// Kernel D: segments 0 and 3 (one context read feeds both)
// ---------------------------------------------------------------------------
__global__ void finalize_kernel(const float* __restrict__ context,
                                const float* __restrict__ q2c,
                                float* __restrict__ out) {
    const size_t i = (size_t)blockIdx.x * 256 + threadIdx.x;   // over B*C*E
    const int e = (int)(i % E);
    const size_t bc = i / E;
    const int b = (int)(bc / C);
    const float cv = context[i];
    const float qv = q2c[b * E + e];
    float* orow = out + bc * (4 * E);
    orow[e]         = cv;        // segment 0: context copy
    orow[3 * E + e] = cv * qv;   // segment 3: ctx * q2c
}

// ---------------------------------------------------------------------------
extern "C" void kernel_launch(void* const* d_in, const int* in_sizes, int n_in,
                              void* d_out, int out_size, void* d_ws, size_t ws_size,
                              hipStream_t stream) {
    const float* context    = (const float*)d_in[0];
    const float* question   = (const float*)d_in[1];
    // d_in[2] = context_mask (bool) — unused by the reference computation
    const float* w_question = (const float*)d_in[3];
    const float* w_context  = (const float*)d_in[4];
    const float* w_multiple = (const float*)d_in[5];
    float* out = (float*)d_out;

    float* wsf  = (float*)d_ws;
    float* qw   = wsf;                          // Bb*Q
    float* mrow = qw + Bb * Q;                  // Bb*C
    float* q2c  = mrow + (size_t)Bb * C;        // Bb*E   (total ws use ~160 KB)

    qw_kernel<<<dim3(Bb), dim3(Q), 0, stream>>>(question, w_question, qw);

    constexpr size_t smem_main =
        (size_t)(Q * QPAD + WAVES * 16 * SPAD + E + E + Q + WAVES * 16) * sizeof(float);
    bidaf_main<<<dim3(C / ROWS_PER_WG, Bb), dim3(WAVES * 32), smem_main, stream>>>(
        context, question, w_context, w_multiple, qw, mrow, out);

    q2c_kernel<<<dim3(Bb), dim3(256), 0, stream>>>(context, mrow, q2c);

    finalize_kernel<<<dim3((Bb * C * E) / 256), dim3(256), 0, stream>>>(context, q2c, out);
}